// vector_quantizer_89833535963913
// MI455X (gfx1250) — compile-verified
//
#include <hip/hip_runtime.h>

// Soft vector quantization for MI455X (gfx1250, wave32).
// B=8, FLAT=262144, K=256, D=4  ->  N = 524288 vectors of dim 4, 256 centroids.
//
// Per 16-vector tile (one wave), single pass:
//   S_t[16x16] = X[16x4] * C_t^T[4x16]   via V_WMMA_F32_16X16X4_F32 (16 tiles)
//   e = exp(2S - ||c||^2) = exp2(S*2log2e) * wt,  wt = exp2(-log2e*||c||^2)
//   (softmax is shift-invariant: -||x||^2 cancels; no row-max needed since
//    |2x.c| <~ 54 for N(0,1) data -> exp stays well inside fp32 range)
//   den += e ; acc += e * c   (the smx*C einsum, full-rate VALU FMA)
//   row reduce via ds_swizzle XOR butterfly over 16-lane halves, store.
//
// Centers are wave-invariant: B operands, wt, and pre-scaled centers cw=wt*c
// are loaded once per wave and held in registers across a grid-stride tile loop.

typedef __attribute__((ext_vector_type(2))) float v2f;
typedef __attribute__((ext_vector_type(4))) float v4f;
typedef __attribute__((ext_vector_type(8))) float v8f;

template <int IMM>
__device__ __forceinline__ float swz(float v) {
  return __int_as_float(__builtin_amdgcn_ds_swizzle(__float_as_int(v), IMM));
}

// XOR-butterfly sum within each 16-lane half (masks 1,2,4,8 never cross the
// lane-16 boundary, matching the WMMA C/D row split: rows j / j+8).
__device__ __forceinline__ float redsum16(float v) {
  v += swz<0x041F>(v);  // xor 1
  v += swz<0x081F>(v);  // xor 2
  v += swz<0x101F>(v);  // xor 4
  v += swz<0x201F>(v);  // xor 8
  return v;
}

__global__ __launch_bounds__(256) void softvq_wmma_kernel(
    const float* __restrict__ x, const float* __restrict__ center,
    float* __restrict__ out, int n_tiles) {
  const int lane  = threadIdx.x & 31;
  const int wave  = threadIdx.x >> 5;
  const int wgrid = gridDim.x * (blockDim.x >> 5);          // total waves
  const int wid   = blockIdx.x * (blockDim.x >> 5) + wave;  // this wave

  const int r  = lane & 15;  // row index (A) / column index (B, C/D)
  const int hi = lane >> 4;  // lane half: K-pair select (A/B), row+8 (C/D)

  const float LOG2E = 1.4426950408889634f;
  const float C1    = 2.0f * LOG2E;  // exp(2S) = exp2(S*C1)

  // ---- Wave-invariant centroid state (held in registers for all tiles).
  v2f   b[16];   // B operands: 4x16 tiles of C^T
  float wt[16];  // exp2(-log2e * ||c_n||^2) for this lane's column n
  v4f   cw[16];  // wt * c_n  (pre-scaled centers for the output einsum)
#pragma unroll
  for (int t = 0; t < 16; ++t) {
    const int n = t * 16 + r;  // centroid column handled by this lane
    b[t] = *(const v2f*)(center + (size_t)n * 4 + hi * 2);
    v4f c4 = *(const v4f*)(center + (size_t)n * 4);
    float csq = c4.x * c4.x + c4.y * c4.y + c4.z * c4.z + c4.w * c4.w;
    float w   = __builtin_amdgcn_exp2f(-LOG2E * csq);
    wt[t] = w;
    cw[t] = (v4f){w * c4.x, w * c4.y, w * c4.z, w * c4.w};
  }

  // ---- Grid-stride over 16-vector tiles (condition is wave-uniform: EXEC
  //      is all-1s at every WMMA).
  for (int tile = wid; tile < n_tiles; tile += wgrid) {
    const int m0 = tile * 16;

    // A operand: 16x4 f32 tile of x.
    // Lanes 0-15 hold {K=0,K=1}, lanes 16-31 hold {K=2,K=3} for row r.
    v2f a = *(const v2f*)(x + (size_t)(m0 + r) * 4 + hi * 2);
    // Speculative prefetch of the next tile's A slice (OOB is dropped).
    __builtin_prefetch(x + ((size_t)m0 + (size_t)wgrid * 16 + r) * 4, 0, 1);

    float den[8];
    v4f   acc[8];
#pragma unroll
    for (int j = 0; j < 8; ++j) {
      den[j] = 0.0f;
      acc[j] = (v4f){0.0f, 0.0f, 0.0f, 0.0f};
    }

#pragma unroll
    for (int t = 0; t < 16; ++t) {
      v8f cz = {};
      v8f S  = __builtin_amdgcn_wmma_f32_16x16x4_f32(
          /*neg_a=*/false, a, /*neg_b=*/false, b[t],
          /*c_mod=*/(short)0, cz, /*reuse_a=*/false, /*reuse_b=*/false);
#pragma unroll
      for (int j = 0; j < 8; ++j) {
        float eu = __builtin_amdgcn_exp2f(S[j] * C1);
        den[j]   = __builtin_fmaf(eu, wt[t], den[j]);
        acc[j].x = __builtin_fmaf(eu, cw[t].x, acc[j].x);
        acc[j].y = __builtin_fmaf(eu, cw[t].y, acc[j].y);
        acc[j].z = __builtin_fmaf(eu, cw[t].z, acc[j].z);
        acc[j].w = __builtin_fmaf(eu, cw[t].w, acc[j].w);
      }
    }

#pragma unroll
    for (int j = 0; j < 8; ++j) {
      den[j]   = redsum16(den[j]);
      acc[j].x = redsum16(acc[j].x);
      acc[j].y = redsum16(acc[j].y);
      acc[j].z = redsum16(acc[j].z);
      acc[j].w = redsum16(acc[j].w);
    }

    // Store: lane 0 writes rows 0..7, lane 16 writes rows 8..15 (float4 each).
    if (r == 0) {
#pragma unroll
      for (int j = 0; j < 8; ++j) {
        const int   row = j + hi * 8;
        const float inv = 1.0f / den[j];
        v4f o = {acc[j].x * inv, acc[j].y * inv, acc[j].z * inv, acc[j].w * inv};
        *(v4f*)(out + (size_t)(m0 + row) * 4) = o;
      }
    }
  }
}

extern "C" void kernel_launch(void* const* d_in, const int* in_sizes, int n_in,
                              void* d_out, int out_size, void* d_ws,
                              size_t ws_size, hipStream_t stream) {
  const float* x      = (const float*)d_in[0];  // [B, FLAT] f32
  const float* center = (const float*)d_in[1];  // [K, D]    f32
  float*       out    = (float*)d_out;          // [B, FLAT] f32

  const int n_vec   = in_sizes[0] / 4;  // D = 4
  const int n_tiles = n_vec / 16;       // 16 vectors per wave tile
  const int threads = 256;              // 8 waves per block
  const int wpb     = threads / 32;

  int blocks = (n_tiles + wpb - 1) / wpb;
  if (blocks > 2048) blocks = 2048;  // grid-stride: ~2 tiles per wave

  softvq_wmma_kernel<<<blocks, threads, 0, stream>>>(x, center, out, n_tiles);
}